// GNN_modules_concat_28879360098865
// MI455X (gfx1250) — compile-verified
//
#include <hip/hip_runtime.h>
#include <hip/hip_bf16.h>
#include <cstdint>

typedef __attribute__((ext_vector_type(2))) float v2f;
typedef __attribute__((ext_vector_type(8))) float v8f;

#define FD 64   // feature dim (all layers 64 -> 64)

// ---------------- degree / symmetric norm ----------------
__global__ void k_deg_init(float* deg, int n) {
    int i = blockIdx.x * blockDim.x + threadIdx.x;
    if (i < n) deg[i] = 1.0f;                 // self loop contributes 1
}

__global__ void k_deg_edges(const int* __restrict__ dst, float* deg, int e) {
    int i = blockIdx.x * blockDim.x + threadIdx.x;
    if (i < e) atomicAdd(&deg[dst[i]], 1.0f);
}

__global__ void k_dinv(float* deg, int n) {
    int i = blockIdx.x * blockDim.x + threadIdx.x;
    if (i < n) deg[i] = rsqrtf(deg[i]);       // deg >= 1 always
}

// ---------------- lin = h @ W  (N x 64 @ 64 x 64), FP32 WMMA ----------------
// One wave computes a 16-row x 64-col tile. W staged in LDS (16 KB) in a
// row-pair-interleaved layout so each B fragment is ONE aligned ds_load_b64:
//   sW[(k>>1)*128 + c*2 + (k&1)] = W[k][c]
// K outer / col-block inner: 4 independent WMMA accumulation chains.
__global__ void k_gemm64(const float* __restrict__ A, const float* __restrict__ W,
                         float* __restrict__ Out, int n) {
    __shared__ float sW[FD * FD];
    for (int idx = threadIdx.x; idx < FD * FD; idx += blockDim.x) {
        const int k = idx >> 6;
        const int c = idx & 63;
        sW[(k >> 1) * 128 + c * 2 + (k & 1)] = W[idx];
    }
    __syncthreads();                          // all waves reach this barrier

    const int lane  = threadIdx.x & 31;
    const int wave  = threadIdx.x >> 5;
    const int tiles = (n + 15) >> 4;
    const int tile  = blockIdx.x * (blockDim.x >> 5) + wave;
    if (tile >= tiles) return;                // wave-uniform guard: EXEC all-1 below

    const int row0  = tile << 4;
    const int m     = lane & 15;              // M index (A) / N index (B,C,D)
    const int khalf = lane >> 4;              // 0: K=0,1   1: K=2,3  (per 4-wide step)
    const int row   = row0 + m;
    const int arow  = (row < n) ? row : (n - 1);   // clamp so EXEC stays all-1

#if defined(__gfx1250__) && __has_builtin(__builtin_amdgcn_wmma_f32_16x16x4_f32)
    // A fragments: lane (khalf*16+m) holds A[row0+m][ks*4 + khalf*2 + j], j=0,1
    v2f a[16];
    const float* ap = A + (size_t)arow * FD + khalf * 2;
#pragma unroll
    for (int ks = 0; ks < 16; ++ks)
        a[ks] = *(const v2f*)(ap + ks * 4);

    v8f acc[4];
#pragma unroll
    for (int nb = 0; nb < 4; ++nb)
        acc[nb] = (v8f){0.f, 0.f, 0.f, 0.f, 0.f, 0.f, 0.f, 0.f};

    // B fragment for (ks, nb): float2 at sW + ks*256 + khalf*128 + nb*32 + m*2
    const float* wbase = &sW[khalf * 128 + m * 2];
#pragma unroll
    for (int ks = 0; ks < 16; ++ks) {         // K = 64 in steps of 4
        const float* wp = wbase + ks * 256;
#pragma unroll
        for (int nb = 0; nb < 4; ++nb) {      // 4 independent acc chains per K-step
            v2f b = *(const v2f*)(wp + nb * 32);
            acc[nb] = __builtin_amdgcn_wmma_f32_16x16x4_f32(
                false, a[ks], false, b, (short)0, acc[nb], false, false);
        }
    }

    // C/D layout: lanes 0-15 -> M=r, lanes 16-31 -> M=8+r; N = lane&15
    if (row0 + 16 <= n) {
        // Fast path (all tiles but possibly the last): unguarded stores,
        // constant-offset immediates from one base pointer.
        float* outp = Out + (size_t)(row0 + khalf * 8) * FD + m;
#pragma unroll
        for (int nb = 0; nb < 4; ++nb) {
#pragma unroll
            for (int r = 0; r < 8; ++r)
                outp[(size_t)r * FD + nb * 16] = acc[nb][r];
        }
    } else {
        // Boundary tile: per-row guard
#pragma unroll
        for (int nb = 0; nb < 4; ++nb) {
            const int n_out = nb * 16 + m;
#pragma unroll
            for (int r = 0; r < 8; ++r) {
                const int m_out = row0 + khalf * 8 + r;
                if (m_out < n) Out[(size_t)m_out * FD + n_out] = acc[nb][r];
            }
        }
    }
#else
    // Scalar fallback (also taken by the host compilation pass)
    if (lane < 16 && row < n) {
        for (int c = 0; c < FD; ++c) {
            float s = 0.f;
            for (int k = 0; k < FD; ++k)
                s += A[(size_t)arow * FD + k] * sW[(k >> 1) * 128 + c * 2 + (k & 1)];
            Out[(size_t)row * FD + c] = s;
        }
    }
#endif
}

// ---------------- agg init with self-loop contribution ----------------
__global__ void k_selfloop(const float* __restrict__ lin, const float* __restrict__ dinv,
                           float* __restrict__ agg, int n) {
    size_t idx = (size_t)blockIdx.x * blockDim.x + threadIdx.x;
    if (idx < (size_t)n * FD) {
        int i = (int)(idx >> 6);
        float dv = dinv[i];
        agg[idx] = lin[idx] * dv * dv;        // self-loop norm = dinv[i]^2
    }
}

// ---------------- edge scatter: one wave32 per edge, lane owns 2 floats ----------------
__global__ void k_scatter(const float* __restrict__ lin, const float* __restrict__ dinv,
                          const int* __restrict__ src, const int* __restrict__ dst,
                          float* __restrict__ agg, int e) {
    size_t t  = (size_t)blockIdx.x * blockDim.x + threadIdx.x;
    int edge  = (int)(t >> 5);
    int lane  = (int)(t & 31);
    if (edge >= e) return;
    int s = src[edge];
    int d = dst[edge];
    float nrm = dinv[s] * dinv[d];
    v2f v = *(const v2f*)(lin + (size_t)s * FD + lane * 2);   // 256B coalesced gather
    float* o = agg + (size_t)d * FD + lane * 2;               // 256B contiguous atomics
    atomicAdd(o + 0, v.x * nrm);
    atomicAdd(o + 1, v.y * nrm);
}

// ---------------- bias + ELU -> h (for next layer) and concat output slice ----------------
__global__ void k_bias_elu(const float* __restrict__ agg, const float* __restrict__ bias,
                           float* __restrict__ h, float* __restrict__ out,
                           int n, int layer_off) {
    size_t idx = (size_t)blockIdx.x * blockDim.x + threadIdx.x;
    if (idx < (size_t)n * FD) {
        int i = (int)(idx >> 6);
        int c = (int)(idx & 63);
        float v  = agg[idx] + bias[c];
        float ev = (v > 0.f) ? v : (expf(v) - 1.f);           // ELU, alpha=1
        h[idx] = ev;
        out[(size_t)i * (3 * FD) + layer_off + c] = ev;       // concat along axis=1
    }
}

extern "C" void kernel_launch(void* const* d_in, const int* in_sizes, int n_in,
                              void* d_out, int out_size, void* d_ws, size_t ws_size,
                              hipStream_t stream) {
    // setup_inputs order: x, edge_index, W1, b1, W2, b2, W3, b3
    const float* x   = (const float*)d_in[0];
    const int*   ei  = (const int*)d_in[1];
    const float* Wl[3] = {(const float*)d_in[2], (const float*)d_in[4], (const float*)d_in[6]};
    const float* bl[3] = {(const float*)d_in[3], (const float*)d_in[5], (const float*)d_in[7]};

    const int n = in_sizes[0] / FD;           // 50000
    const int e = in_sizes[1] / 2;            // 800000
    const int* src = ei;                      // edge_index[0], row-major flat
    const int* dst = ei + e;                  // edge_index[1]

    // workspace layout (floats): dinv[n] | h[n*64] | lin[n*64] | agg[n*64]  (~38.6 MB)
    float* ws   = (float*)d_ws;
    float* dinv = ws;
    float* h    = dinv + n;
    float* lin  = h   + (size_t)n * FD;
    float* agg  = lin + (size_t)n * FD;

    const int B = 256;
    const int gN  = (n + B - 1) / B;
    const int gE  = (e + B - 1) / B;
    const int tiles = (n + 15) / 16;
    const int gGemm = (tiles + (B / 32) - 1) / (B / 32);
    const size_t nd = (size_t)n * FD;
    const int gElem = (int)((nd + B - 1) / B);
    const int gScat = (int)(((size_t)e * 32 + B - 1) / B);

    // degree (with self loops) -> dinv, in place
    k_deg_init <<<gN, B, 0, stream>>>(dinv, n);
    k_deg_edges<<<gE, B, 0, stream>>>(dst, dinv, e);
    k_dinv     <<<gN, B, 0, stream>>>(dinv, n);

    const float* hin = x;
    for (int l = 0; l < 3; ++l) {
        k_gemm64  <<<gGemm, B, 0, stream>>>(hin, Wl[l], lin, n);
        k_selfloop<<<gElem, B, 0, stream>>>(lin, dinv, agg, n);
        k_scatter <<<gScat, B, 0, stream>>>(lin, dinv, src, dst, agg, e);
        k_bias_elu<<<gElem, B, 0, stream>>>(agg, bl[l], h, (float*)d_out, n, l * FD);
        hin = h;
    }
}